// DrugInteractionGNN_22471268892879
// MI455X (gfx1250) — compile-verified
//
#include <hip/hip_runtime.h>
#include <hip/hip_bf16.h>

typedef __attribute__((ext_vector_type(2))) float v2f;
typedef __attribute__((ext_vector_type(8))) float v8f;

// ---------------------------------------------------------------------------
// Degree / normalization kernels
// ---------------------------------------------------------------------------
__global__ __launch_bounds__(256) void init_deg_kernel(float* __restrict__ deg, int n) {
    int i = blockIdx.x * blockDim.x + threadIdx.x;
    if (i < n) deg[i] = 1.0f;               // self-loop contributes 1
}

__global__ __launch_bounds__(256) void accum_deg_kernel(const int* __restrict__ dst,
                                                        float* __restrict__ deg, int e) {
    int i = blockIdx.x * blockDim.x + threadIdx.x;
    if (i < e) unsafeAtomicAdd(&deg[dst[i]], 1.0f);
}

__global__ __launch_bounds__(256) void dinv_kernel(float* __restrict__ deg, int n) {
    int i = blockIdx.x * blockDim.x + threadIdx.x;
    if (i < n) {
        float d = deg[i];
        deg[i] = (d > 0.0f) ? (1.0f / sqrtf(d)) : 0.0f;
    }
}

// ---------------------------------------------------------------------------
// WMMA f32 GEMM:  C[M x N] = op(A)[M x K] @ W[K x N]   (op = optional ReLU)
// Block = 256 threads = 8 waves; each wave computes a 16-row strip across all
// N columns using V_WMMA_F32_16X16X4_F32.  W staged through 32KB LDS chunks.
// ---------------------------------------------------------------------------
template <int K, int N, bool RELU>
__global__ __launch_bounds__(256) void gemm_wmma_kernel(const float* __restrict__ A,
                                                        const float* __restrict__ W,
                                                        float* __restrict__ C, int M) {
    constexpr int KC = 8192 / N;          // K-chunk so chunk = 8192 floats = 32KB
    static_assert(K % KC == 0, "K must be multiple of chunk");
    constexpr int NT = N / 16;            // 16x16 output tiles per wave strip
    __shared__ float lw[8192];            // KC x N chunk of W

    const int tid  = threadIdx.x;
    const int wave = tid >> 5;
    const int lane = tid & 31;
    const int row0 = (blockIdx.x * 8 + wave) * 16;
    const bool active = (row0 < M);       // wave-uniform: EXEC stays all-1s

    const int m    = lane & 15;           // row-in-tile (A) / col-in-tile (B)
    const int hi   = lane >> 4;           // half-wave select
    const int koff = hi * 2;              // K sub-offset per ISA 32-bit A layout

    v8f zero = {};
    v8f acc[NT];
#pragma unroll
    for (int t = 0; t < NT; ++t) acc[t] = zero;

    const float* arow = A + (size_t)(row0 + m) * K + koff;

    for (int kc = 0; kc < K; kc += KC) {
        __syncthreads();
        // cooperative 32KB chunk load of W rows [kc, kc+KC)
        const float* wsrc = W + (size_t)kc * N;
#pragma unroll
        for (int idx = tid * 4; idx < KC * N; idx += 256 * 4) {
            *(float4*)&lw[idx] = *(const float4*)&wsrc[idx];
        }
        __syncthreads();

        if (active) {
#pragma unroll 4
            for (int k = 0; k < KC; k += 4) {
                float2 av = *(const float2*)(arow + kc + k);
                if (RELU) { av.x = fmaxf(av.x, 0.0f); av.y = fmaxf(av.y, 0.0f); }
                v2f a; a.x = av.x; a.y = av.y;
#pragma unroll
                for (int t = 0; t < NT; ++t) {
                    const int n = t * 16 + m;
                    v2f b;
                    b.x = lw[(k + koff) * N + n];
                    b.y = lw[(k + koff + 1) * N + n];
                    acc[t] = __builtin_amdgcn_wmma_f32_16x16x4_f32(
                        /*neg_a=*/false, a, /*neg_b=*/false, b,
                        /*c_mod=*/(short)0, acc[t],
                        /*reuse_a=*/false, /*reuse_b=*/false);
                }
            }
        }
    }

    if (active) {
        // C/D layout: VGPR r -> row (r + 8*hi), col = lane&15 within tile
#pragma unroll
        for (int t = 0; t < NT; ++t) {
            const int col = t * 16 + m;
#pragma unroll
            for (int r = 0; r < 8; ++r) {
                const int row = row0 + r + hi * 8;
                C[(size_t)row * N + col] = acc[t][r];
            }
        }
    }
}

// ---------------------------------------------------------------------------
// Aggregation: out = h * dinv^2 + bias  (self-loop + bias), float4 vectorized
// ---------------------------------------------------------------------------
template <int F>
__global__ __launch_bounds__(256) void agg_init_kernel(const float* __restrict__ h,
                                                       const float* __restrict__ dinv,
                                                       const float* __restrict__ bias,
                                                       float* __restrict__ out, int n) {
    constexpr int F4 = F / 4;
    int idx = blockIdx.x * blockDim.x + threadIdx.x;   // over n * F4
    if (idx >= n * F4) return;
    int i  = idx / F4;
    int f4 = idx - i * F4;
    float di = dinv[i];
    float s  = di * di;
    float4 hv = ((const float4*)h)[idx];
    float4 bv = ((const float4*)bias)[f4];
    float4 o;
    o.x = hv.x * s + bv.x;
    o.y = hv.y * s + bv.y;
    o.z = hv.z * s + bv.z;
    o.w = hv.w * s + bv.w;
    ((float4*)out)[idx] = o;
}

// ---------------------------------------------------------------------------
// Edge scatter: out[dst] += h[src] * dinv[src]*dinv[dst], feature-parallel.
// Coalesced gather of h[src], hardware f32 atomics into L2-resident out.
// ---------------------------------------------------------------------------
template <int F>
__global__ __launch_bounds__(256) void agg_edge_kernel(const float* __restrict__ h,
                                                       const float* __restrict__ dinv,
                                                       const int* __restrict__ src,
                                                       const int* __restrict__ dst,
                                                       float* __restrict__ out, int e) {
    constexpr int EPB = 256 / F;
    int eidx = blockIdx.x * EPB + (threadIdx.x / F);
    if (eidx >= e) return;
    int f = threadIdx.x & (F - 1);
    int s = src[eidx];
    int d = dst[eidx];
    float norm = dinv[s] * dinv[d];
    float v = h[(size_t)s * F + f] * norm;
    unsafeAtomicAdd(&out[(size_t)d * F + f], v);
}

// ---------------------------------------------------------------------------
// Host driver
// ---------------------------------------------------------------------------
extern "C" void kernel_launch(void* const* d_in, const int* in_sizes, int n_in,
                              void* d_out, int out_size, void* d_ws, size_t ws_size,
                              hipStream_t stream) {
    const float* x  = (const float*)d_in[0];
    const int*   ei = (const int*)d_in[1];
    const float* W1 = (const float*)d_in[2];
    const float* b1 = (const float*)d_in[3];
    const float* W2 = (const float*)d_in[4];
    const float* b2 = (const float*)d_in[5];
    float* out = (float*)d_out;

    const int h1f  = in_sizes[3];            // 128
    const int h2f  = in_sizes[5];            // 64
    const int inf  = in_sizes[2] / h1f;      // 256
    const int N    = in_sizes[0] / inf;      // 100000
    const int E    = in_sizes[1] / 2;        // 1600000
    const int* src = ei;
    const int* dst = ei + E;

    // workspace layout (256B aligned slots)
    char* ws = (char*)d_ws;
    size_t off = 0;
    auto carve = [&](size_t bytes) -> void* {
        void* p = ws + off;
        off = (off + bytes + 255) & ~(size_t)255;
        return p;
    };
    float* dinv = (float*)carve((size_t)N * sizeof(float));
    float* h1   = (float*)carve((size_t)N * h1f * sizeof(float));
    float* g1   = (float*)carve((size_t)N * h1f * sizeof(float));
    float* h2   = (float*)carve((size_t)N * h2f * sizeof(float));
    (void)ws_size; (void)n_in; (void)out_size;

    const int T = 256;
    // 1) symmetric degree normalization
    init_deg_kernel<<<(N + T - 1) / T, T, 0, stream>>>(dinv, N);
    accum_deg_kernel<<<(E + T - 1) / T, T, 0, stream>>>(dst, dinv, E);
    dinv_kernel<<<(N + T - 1) / T, T, 0, stream>>>(dinv, N);

    // 2) layer 1: h1 = x @ W1  (WMMA f32), then aggregate into g1
    gemm_wmma_kernel<256, 128, false><<<(N + 127) / 128, T, 0, stream>>>(x, W1, h1, N);
    agg_init_kernel<128><<<((N * 32) + T - 1) / T, T, 0, stream>>>(h1, dinv, b1, g1, N);
    agg_edge_kernel<128><<<(E + 1) / 2, T, 0, stream>>>(h1, dinv, src, dst, g1, E);

    // 3) layer 2: h2 = relu(g1) @ W2 (ReLU fused into A load), aggregate to out
    gemm_wmma_kernel<128, 64, true><<<(N + 127) / 128, T, 0, stream>>>(g1, W2, h2, N);
    agg_init_kernel<64><<<((N * 16) + T - 1) / T, T, 0, stream>>>(h2, dinv, b2, out, N);
    agg_edge_kernel<64><<<(E + 3) / 4, T, 0, stream>>>(h2, dinv, src, dst, out, E);
}